// Transformer_38809324486825
// MI455X (gfx1250) — compile-verified
//
#include <hip/hip_runtime.h>
#include <hip/hip_bf16.h>
#include <math.h>

typedef __attribute__((ext_vector_type(2))) float v2f;
typedef __attribute__((ext_vector_type(4))) float v4f;
typedef __attribute__((ext_vector_type(8))) float v8f;
typedef __attribute__((__vector_size__(16))) int v4i;

#define THRESH_V 0.05f
#define EPS_V 1e-5f

#if defined(__HIP_DEVICE_COMPILE__) && __has_builtin(__builtin_amdgcn_global_load_async_to_lds_b128)
#define HAS_ASYNC_LDS 1
#endif

// Address-space-qualified int4 pointers for the async-LDS builtin:
// param0 = global (addrspace(1), prints as __device__), param1 = LDS (addrspace(3)).
typedef __attribute__((address_space(1))) v4i as1_v4i;
typedef __attribute__((address_space(3))) v4i as3_v4i;

__device__ __forceinline__ float thresh_f(float v) {
    return fabsf(v) > THRESH_V ? v : 0.0f;
}

// Copy 16 bytes global -> LDS. Async (ASYNCcnt-tracked DMA) when available,
// else synchronous NT load + LDS store.
__device__ __forceinline__ void copy16_g2l(const float* __restrict__ g,
                                           float* __restrict__ l) {
#ifdef HAS_ASYNC_LDS
    __builtin_amdgcn_global_load_async_to_lds_b128((as1_v4i*)g, (as3_v4i*)l, 0, 0);
#else
    *(v4f*)l = __builtin_nontemporal_load((const v4f*)g);
#endif
}

__device__ __forceinline__ void wait_async_zero() {
#ifdef HAS_ASYNC_LDS
#if __has_builtin(__builtin_amdgcn_s_wait_asynccnt)
    __builtin_amdgcn_s_wait_asynccnt(0);
#else
    asm volatile("s_wait_asynccnt 0x0" ::: "memory");
#endif
#endif
}

// ---------------------------------------------------------------------------
// WMMA-based GEMV core (one block = 256 threads = 8 waves; 128 rows/block).
// K processed in 64-wide tiles, double-buffered through LDS:
//   - 128x64 fp32 weight tile staged with coalesced 16B async copies
//     (row stride padded to 68 floats -> conflict-free b64 reads)
//   - 16x V_WMMA_F32_16X16X4_F32 per tile: A = 16 rows x 4 K of weights
//     (bandwidth-heavy operand at 100% utilization), B = x chunk in column 0.
//   - two independent accumulators break the WMMA dependency chain.
// D[m][0] lives in lane 0 (m=0..7 -> VGPR m) and lane 16 (m=8..15).
// ---------------------------------------------------------------------------
__device__ __forceinline__ void stage_tile(const float* __restrict__ Wbase,
                                           int K, int kb, float* __restrict__ buf,
                                           int tid, int kLen)
{
#pragma unroll
    for (int j = 0; j < 8; ++j) {
        int f  = j * 256 + tid;     // float4 index: 2048 over 256 threads
        int r  = f >> 4;            // row 0..127
        int c4 = f & 15;            // float4 column 0..15
        const float* gp = Wbase + (size_t)r * K + kb + c4 * 4;
        copy16_g2l(gp, &buf[r * 68 + c4 * 4]);
        if (j == 0 && kb + 64 < kLen)
            __builtin_prefetch((const void*)(gp + 64), 0, 1);  // tile after next
    }
}

__device__ __forceinline__ void gemv_block_wmma(
    const float* __restrict__ W,    // N x K row-major
    float* __restrict__ tile,       // LDS [2 * 128*68] (double buffer)
    const float* __restrict__ xv,   // LDS x slice [kLen]
    float* __restrict__ y,
    int rowBlock, int K, int k0, int kLen)
{
    const int tid  = threadIdx.x;
    const int wv   = tid >> 5;
    const int lane = tid & 31;
    const int half = lane >> 4;   // 0: lanes 0-15, 1: lanes 16-31
    const int lr   = lane & 15;

    const float* Wbase = W + (size_t)rowBlock * K + k0;
    float* buf0 = tile;
    float* buf1 = tile + 128 * 68;

    const int nt = kLen >> 6;     // number of 64-wide tiles
    stage_tile(Wbase, K, 0, buf0, tid, kLen);

    v8f acc0 = {}, acc1 = {};
    for (int t = 0; t < nt; ++t) {
        float* cur = (t & 1) ? buf1 : buf0;
        float* nxt = (t & 1) ? buf0 : buf1;

        wait_async_zero();        // own tile-t copies landed
        __syncthreads();          // everyone's tile-t copies visible;
                                  // everyone done computing tile t-1 from nxt
        if (t + 1 < nt)
            stage_tile(Wbase, K, (t + 1) * 64, nxt, tid, kLen);

        const float* tw = &cur[(wv * 16 + lr) * 68 + 2 * half];
        const float* tx = &xv[t * 64 + 2 * half];
#pragma unroll
        for (int i = 0; i < 16; i += 2) {
            // A (16x4 f32): lanes 0-15 hold K=0,1 ; lanes 16-31 hold K=2,3.
            v2f a0 = *(const v2f*)(tw + 4 * i);
            v2f a1 = *(const v2f*)(tw + 4 * i + 4);
            // B (4x16 f32), column 0 only: VGPR0 = {K0|K2}, VGPR1 = {K1|K3}.
            v2f b0, b1;
            b0.x = (lr == 0) ? tx[4 * i]     : 0.0f;
            b0.y = (lr == 0) ? tx[4 * i + 1] : 0.0f;
            b1.x = (lr == 0) ? tx[4 * i + 4] : 0.0f;
            b1.y = (lr == 0) ? tx[4 * i + 5] : 0.0f;
            acc0 = __builtin_amdgcn_wmma_f32_16x16x4_f32(
                false, a0, false, b0, (short)0, acc0, false, false);
            acc1 = __builtin_amdgcn_wmma_f32_16x16x4_f32(
                false, a1, false, b1, (short)0, acc1, false, false);
        }
    }

    // Extract column 0 of D: lane 0 has rows +0..7, lane 16 has rows +8..15.
    if (lr == 0) {
        int rb = rowBlock + wv * 16 + half * 8;
#pragma unroll
        for (int r = 0; r < 8; ++r) y[rb + r] = acc0[r] + acc1[r];
    }
}

__global__ void __launch_bounds__(256) gemv_wmma_k(
    const float* __restrict__ W, const float* __restrict__ x,
    float* __restrict__ y, int K, int kLen, int N)
{
    __shared__ float tile[2 * 128 * 68];
    __shared__ float xv[4096];
    const int k0 = blockIdx.y * kLen;
    for (int i = threadIdx.x; i < kLen; i += 256) xv[i] = x[k0 + i];
    gemv_block_wmma(W, tile, xv, y + (size_t)blockIdx.y * N,
                    blockIdx.x * 128, K, k0, kLen);
}

// Fused w1/w3 GEMV: blocks [0,halfBlocks) -> W1/y1, rest -> W3/y3.
__global__ void __launch_bounds__(256) gemv_wmma_dual_k(
    const float* __restrict__ W1, const float* __restrict__ W3,
    const float* __restrict__ x, float* __restrict__ y1,
    float* __restrict__ y3, int K, int halfBlocks)
{
    __shared__ float tile[2 * 128 * 68];
    __shared__ float xv[4096];
    for (int i = threadIdx.x; i < K; i += 256) xv[i] = x[i];
    const float* W;
    float* y;
    int rb;
    if ((int)blockIdx.x < halfBlocks) {
        W = W1; y = y1; rb = blockIdx.x * 128;
    } else {
        W = W3; y = y3; rb = ((int)blockIdx.x - halfBlocks) * 128;
    }
    gemv_block_wmma(W, tile, xv, y, rb, K, 0, K);
}

// ---------------------------------------------------------------------------
// Elementwise / reduction helper kernels
// ---------------------------------------------------------------------------

// ht = thresh( rmsnorm(x) * w )   (single block, N = 4096)
__global__ void __launch_bounds__(256) rmsnorm_thresh_k(
    const float* __restrict__ x, const float* __restrict__ w,
    float* __restrict__ o, int N)
{
    __shared__ float red[256];
    const int t = threadIdx.x;
    float s = 0.0f;
    for (int i = t; i < N; i += 256) { float v = x[i]; s += v * v; }
    red[t] = s; __syncthreads();
    for (int off = 128; off > 0; off >>= 1) {
        if (t < off) red[t] += red[t + off];
        __syncthreads();
    }
    const float scale = rsqrtf(red[0] / (float)N + EPS_V);
    for (int i = t; i < N; i += 256) {
        float v = x[i] * scale * w[i];
        o[i] = thresh_f(v);
    }
}

// o = thresh( sum over S split-K partials )
__global__ void __launch_bounds__(256) reduceS_thresh_k(
    const float* __restrict__ part, float* __restrict__ o, int N, int S)
{
    int i = blockIdx.x * 256 + threadIdx.x;
    if (i < N) {
        float s = 0.0f;
        for (int j = 0; j < S; ++j) s += part[j * N + i];
        o[i] = thresh_f(s);
    }
}

// h = x + sum(part, S);  hft = thresh( rmsnorm(h) * w )  (single block, N=4096)
__global__ void __launch_bounds__(256) residual_rmsnorm_thresh_k(
    const float* __restrict__ x, const float* __restrict__ part,
    const float* __restrict__ w, float* __restrict__ h,
    float* __restrict__ hft, int N, int S)
{
    __shared__ float red[256];
    const int t = threadIdx.x;
    float hv[16];
    float s = 0.0f;
#pragma unroll
    for (int j = 0; j < 16; ++j) {
        int i = t + j * 256;
        float v = x[i];
        for (int p = 0; p < S; ++p) v += part[p * N + i];
        hv[j] = v;
        h[i] = v;
        s += v * v;
    }
    red[t] = s; __syncthreads();
    for (int off = 128; off > 0; off >>= 1) {
        if (t < off) red[t] += red[t + off];
        __syncthreads();
    }
    const float scale = rsqrtf(red[0] / (float)N + EPS_V);
#pragma unroll
    for (int j = 0; j < 16; ++j) {
        int i = t + j * 256;
        float v = hv[j] * scale * w[i];
        hft[i] = thresh_f(v);
    }
}

// a = thresh( silu(gate) * up )
__global__ void __launch_bounds__(256) silu_mul_thresh_k(
    const float* __restrict__ g, const float* __restrict__ u,
    float* __restrict__ a, int N)
{
    int i = blockIdx.x * 256 + threadIdx.x;
    if (i < N) {
        float gv = g[i];
        float sv = gv / (1.0f + expf(-gv));
        a[i] = thresh_f(sv * u[i]);
    }
}

// out = h + sum over S split-K partials
__global__ void __launch_bounds__(256) final_add_k(
    const float* __restrict__ h, const float* __restrict__ part,
    float* __restrict__ o, int N, int S)
{
    int i = blockIdx.x * 256 + threadIdx.x;
    if (i < N) {
        float s = h[i];
        for (int j = 0; j < S; ++j) s += part[j * N + i];
        o[i] = s;
    }
}

// ---------------------------------------------------------------------------
// Driver
// ---------------------------------------------------------------------------
extern "C" void kernel_launch(void* const* d_in, const int* in_sizes, int n_in,
                              void* d_out, int out_size, void* d_ws, size_t ws_size,
                              hipStream_t stream)
{
    const float* x    = (const float*)d_in[0];
    // d_in[1] = freqs_cis: RoPE at position 0 is identity -> unused
    const float* wqkv = (const float*)d_in[2];
    const float* wo   = (const float*)d_in[3];
    const float* w1   = (const float*)d_in[4];
    const float* w2   = (const float*)d_in[5];
    const float* w3   = (const float*)d_in[6];
    const float* anw  = (const float*)d_in[7];
    const float* fnw  = (const float*)d_in[8];
    // d_in[9] = mask: length-1 softmax axis -> attention output == v, mask dead
    float* out = (float*)d_out;
    float* ws  = (float*)d_ws;

    const int DIMV  = 4096;
    const int INTER = 11008;
    const int SPLIT_SQ = 8;   // 4096-deep GEMVs: kLen = 512 (8 tiles)
    const int SPLIT_W2 = 4;   // 11008-deep GEMV: kLen = 2752 (43 tiles)

    float* ht    = ws;                          // 4096
    float* vpart = ht + DIMV;                   // 8*4096
    float* vt    = vpart + SPLIT_SQ * DIMV;     // 4096
    float* ypart = vt + DIMV;                   // 8*4096
    float* h     = ypart + SPLIT_SQ * DIMV;     // 4096
    float* hft   = h + DIMV;                    // 4096
    float* gate  = hft + DIMV;                  // 11008
    float* up    = gate + INTER;                // 11008
    float* act   = up + INTER;                  // 11008
    float* dpart = act + INTER;                 // 4*4096

    // v projection weights = rows [8192, 12288) of wqkv
    const float* v0 = wqkv + (size_t)2 * DIMV * DIMV;

    // 1) ht = thresh(rmsnorm(x, attn_norm_w))
    rmsnorm_thresh_k<<<1, 256, 0, stream>>>(x, anw, ht, DIMV);
    // 2) v = v0 @ ht   (split-K=8)
    gemv_wmma_k<<<dim3(DIMV / 128, SPLIT_SQ), 256, 0, stream>>>(
        v0, ht, vpart, DIMV, DIMV / SPLIT_SQ, DIMV);
    // 3) vt = thresh(sum partials)   [attention output == v for SEQ=1]
    reduceS_thresh_k<<<DIMV / 256, 256, 0, stream>>>(vpart, vt, DIMV, SPLIT_SQ);
    // 4) y = wo @ vt   (split-K=8)
    gemv_wmma_k<<<dim3(DIMV / 128, SPLIT_SQ), 256, 0, stream>>>(
        wo, vt, ypart, DIMV, DIMV / SPLIT_SQ, DIMV);
    // 5) h = x + y ; hft = thresh(rmsnorm(h, ffn_norm_w))
    residual_rmsnorm_thresh_k<<<1, 256, 0, stream>>>(x, ypart, fnw, h, hft,
                                                     DIMV, SPLIT_SQ);
    // 6) gate = w1 @ hft ; up = w3 @ hft   (fused, 172 blocks)
    gemv_wmma_dual_k<<<2 * (INTER / 128), 256, 0, stream>>>(
        w1, w3, hft, gate, up, DIMV, INTER / 128);
    // 7) act = thresh(silu(gate) * up)
    silu_mul_thresh_k<<<(INTER + 255) / 256, 256, 0, stream>>>(gate, up, act, INTER);
    // 8) down = w2 @ act   (split-K=4, K=11008)
    gemv_wmma_k<<<dim3(DIMV / 128, SPLIT_W2), 256, 0, stream>>>(
        w2, act, dpart, INTER, INTER / SPLIT_W2, DIMV);
    // 9) out = h + down
    final_add_k<<<DIMV / 256, 256, 0, stream>>>(h, dpart, out, DIMV, SPLIT_W2);
}